// RAT_2422361555377
// MI455X (gfx1250) — compile-verified
//
#include <hip/hip_runtime.h>
#include <hip/hip_bf16.h>

#define DCH 128

typedef float v2f __attribute__((ext_vector_type(2)));
typedef float v8f __attribute__((ext_vector_type(8)));

// ---------------- degree / normalization ----------------
__global__ void deg_init_kernel(float* deg, int n) {
    int i = blockIdx.x * blockDim.x + threadIdx.x;
    if (i < n) deg[i] = 1.0f;  // self-loop contributes 1
}

__global__ void deg_count_kernel(float* deg, const int* __restrict__ dst, int e) {
    int i = blockIdx.x * blockDim.x + threadIdx.x;
    if (i < e) atomicAdd(&deg[dst[i]], 1.0f);
}

__global__ void rsqrt_kernel(float* deg, int n) {
    int i = blockIdx.x * blockDim.x + threadIdx.x;
    if (i < n) deg[i] = rsqrtf(deg[i]);
}

// ---------------- fp32 WMMA GEMM: H = (RELU?)A @ W ----------------
// A: [n, 128] row-major, W: [128,128] row-major, H: [n,128].
// Block: 256 threads = 8 waves; block covers 16 rows, wave w covers cols 16w..16w+15.
// RELU is a compile-time flag so layer 1 has zero overhead and layers 2/3 get a
// single v_max_num_f32 per element (co-executes with the WMMA pipe).
template <bool RELU>
__global__ void __launch_bounds__(256)
gemm_wmma_f32_kernel(const float* __restrict__ A, const float* __restrict__ W,
                     float* __restrict__ H, int n) {
    const int lane = threadIdx.x & 31;
    const int wave = threadIdx.x >> 5;
    const int half = lane >> 4;        // selects K-pair (ISA A/B fragment layout)
    const int mn   = lane & 15;        // M for A-frag, N for B-frag
    const int row0 = blockIdx.x * 16;
    const int col0 = wave * 16;

    int arow = row0 + mn;
    if (arow > n - 1) arow = n - 1;    // clamp (n is a multiple of 16 here, but be safe)

    const float* __restrict__ ap = A + (size_t)arow * DCH + 2 * half;          // A[m][k0+2h+{0,1}]
    const float* __restrict__ bp = W + (size_t)(2 * half) * DCH + col0 + mn;   // W[k0+2h+{0,1}][n]

    v8f acc = {};
#pragma unroll
    for (int k0 = 0; k0 < DCH; k0 += 4) {
        v2f a, b;
        a.x = ap[k0];
        a.y = ap[k0 + 1];
        if (RELU) { a.x = fmaxf(a.x, 0.0f); a.y = fmaxf(a.y, 0.0f); }
        b.x = bp[(size_t)k0 * DCH];
        b.y = bp[(size_t)k0 * DCH + DCH];
        acc = __builtin_amdgcn_wmma_f32_16x16x4_f32(
            /*neg_a=*/false, a, /*neg_b=*/false, b,
            /*c_mod=*/(short)0, acc, /*reuse_a=*/false, /*reuse_b=*/false);
    }

#pragma unroll
    for (int j = 0; j < 8; ++j) {
        int r = row0 + j + 8 * half;   // C/D layout: VGPR j -> M = j (lanes 0-15) / j+8 (16-31)
        if (r < n) H[(size_t)r * DCH + col0 + mn] = acc[j];
    }
}

// ---------------- aggregation: self-loop + bias init ----------------
// out[i,:] = h[i,:] * dis[i]^2 + b[:]
__global__ void agg_init_kernel(float* __restrict__ out, const float* __restrict__ h,
                                const float* __restrict__ dis, const float* __restrict__ b,
                                int n) {
    int t = blockIdx.x * blockDim.x + threadIdx.x;   // n*32 threads, 4 ch each
    int i = t >> 5;
    int c = (t & 31) * 4;
    if (i >= n) return;
    float s = dis[i]; s = s * s;
    const float4 hv = *(const float4*)(h + (size_t)i * DCH + c);
    const float4 bv = *(const float4*)(b + c);
    float4 o;
    o.x = hv.x * s + bv.x;
    o.y = hv.y * s + bv.y;
    o.z = hv.z * s + bv.z;
    o.w = hv.w * s + bv.w;
    *(float4*)(out + (size_t)i * DCH + c) = o;
}

// ---------------- aggregation: edge scatter (one wave per edge) ----------------
__global__ void edge_scatter_kernel(float* __restrict__ out, const float* __restrict__ h,
                                    const float* __restrict__ dis,
                                    const int* __restrict__ src, const int* __restrict__ dst,
                                    int e) {
    int t = blockIdx.x * blockDim.x + threadIdx.x;
    int edge = t >> 5;
    int lane = t & 31;
    if (edge >= e) return;
    int s = __builtin_amdgcn_readfirstlane(src[edge]);  // wave-uniform endpoints
    int d = __builtin_amdgcn_readfirstlane(dst[edge]);
    float norm = dis[s] * dis[d];
    int c = lane * 4;
    const float4 v = *(const float4*)(h + (size_t)s * DCH + c);
    float* o = out + (size_t)d * DCH + c;
    atomicAdd(o + 0, v.x * norm);
    atomicAdd(o + 1, v.y * norm);
    atomicAdd(o + 2, v.z * norm);
    atomicAdd(o + 3, v.w * norm);
}

// ---------------- driver ----------------
extern "C" void kernel_launch(void* const* d_in, const int* in_sizes, int n_in,
                              void* d_out, int out_size, void* d_ws, size_t ws_size,
                              hipStream_t stream) {
    const float* x      = (const float*)d_in[0];
    const float* W_role = (const float*)d_in[1];
    const float* b_role = (const float*)d_in[2];
    const float* W2     = (const float*)d_in[3];
    const float* b2     = (const float*)d_in[4];
    const float* W1     = (const float*)d_in[5];
    const float* b1     = (const float*)d_in[6];
    const int*   e_norm = (const int*)d_in[7];
    const int*   e_role = (const int*)d_in[8];

    const int n  = in_sizes[0] / DCH;
    const int En = in_sizes[7] / 2;
    const int Er = in_sizes[8] / 2;
    const int* src_n = e_norm;       const int* dst_n = e_norm + En;
    const int* src_r = e_role;       const int* dst_r = e_role + Er;

    float* h     = (float*)d_ws;                       // n*DCH
    float* bufA  = h + (size_t)n * DCH;                // n*DCH
    float* dis_r = bufA + (size_t)n * DCH;             // n
    float* dis_n = dis_r + n;                          // n
    float* out   = (float*)d_out;

    const int T = 256;
    const int nb_n    = (n + T - 1) / T;
    const int nb_en   = (En + T - 1) / T;
    const int nb_er   = (Er + T - 1) / T;
    const int nb_rows = (n + 15) / 16;
    const int nb_n32  = (int)(((size_t)n * 32 + T - 1) / T);
    const int nb_esn  = (int)(((size_t)En * 32 + T - 1) / T);
    const int nb_esr  = (int)(((size_t)Er * 32 + T - 1) / T);

    // ---- degrees -> dis (rsqrt) for both graphs ----
    deg_init_kernel<<<nb_n, T, 0, stream>>>(dis_r, n);
    deg_init_kernel<<<nb_n, T, 0, stream>>>(dis_n, n);
    deg_count_kernel<<<nb_er, T, 0, stream>>>(dis_r, dst_r, Er);
    deg_count_kernel<<<nb_en, T, 0, stream>>>(dis_n, dst_n, En);
    rsqrt_kernel<<<nb_n, T, 0, stream>>>(dis_r, n);
    rsqrt_kernel<<<nb_n, T, 0, stream>>>(dis_n, n);

    // ---- layer 1: conv_role(x) -> bufA (pre-ReLU; ReLU folded into next GEMM) ----
    gemm_wmma_f32_kernel<false><<<nb_rows, T, 0, stream>>>(x, W_role, h, n);
    agg_init_kernel<<<nb_n32, T, 0, stream>>>(bufA, h, dis_r, b_role, n);
    edge_scatter_kernel<<<nb_esr, T, 0, stream>>>(bufA, h, dis_r, src_r, dst_r, Er);

    // ---- layer 2: conv2(relu(bufA)) -> bufA ----
    gemm_wmma_f32_kernel<true><<<nb_rows, T, 0, stream>>>(bufA, W2, h, n);
    agg_init_kernel<<<nb_n32, T, 0, stream>>>(bufA, h, dis_n, b2, n);
    edge_scatter_kernel<<<nb_esn, T, 0, stream>>>(bufA, h, dis_n, src_n, dst_n, En);

    // ---- layer 3: conv1(relu(bufA)) -> out ----
    gemm_wmma_f32_kernel<true><<<nb_rows, T, 0, stream>>>(bufA, W1, h, n);
    agg_init_kernel<<<nb_n32, T, 0, stream>>>(out, h, dis_n, b1, n);
    edge_scatter_kernel<<<nb_esn, T, 0, stream>>>(out, h, dis_n, src_n, dst_n, En);
}